// Bi_attention_30296699305990
// MI455X (gfx1250) — compile-verified
//
#include <hip/hip_runtime.h>
#include <math.h>

typedef __attribute__((ext_vector_type(2))) float v2f;
typedef __attribute__((ext_vector_type(8))) float v8f;

constexpr int NB = 8;     // batch
constexpr int LC = 512;   // context length
constexpr int LQ = 64;    // question length
constexpr int ND = 256;   // feature dim
constexpr float NEG_BIG = 1e30f;

// ---------------------------------------------------------------------------
// Kernel 1: question-row dual dot products only (context dots are fused into
// the WMMA kernel so context is streamed exactly once).
//   sq = row . w_q ,  q1 = row . w_mem
// One wave32 per row; lane-strided loads + xor-shuffle reduction.
// ---------------------------------------------------------------------------
__global__ void qrow_reduce_kernel(const float* __restrict__ qst,
                                   const float* __restrict__ att_w,
                                   const float* __restrict__ w_mem,
                                   float* __restrict__ sq_ws,
                                   float* __restrict__ q1_ws) {
  const int qrow = (blockIdx.x * blockDim.x + threadIdx.x) >> 5;
  const int lane = threadIdx.x & 31;
  if (qrow >= NB * LQ) return;

  const float* row = qst + (size_t)qrow * ND;
  const float* wq  = att_w + ND;  // w_q = att_w[D:2D]
  float s0 = 0.f, s1 = 0.f;
  #pragma unroll
  for (int k = lane; k < ND; k += 32) {
    float x = row[k];
    s0 += x * wq[k];
    s1 += x * w_mem[k];
  }
  #pragma unroll
  for (int off = 16; off > 0; off >>= 1) {
    s0 += __shfl_xor(s0, off, 32);
    s1 += __shfl_xor(s1, off, 32);
  }
  if (lane == 0) { sq_ws[qrow] = s0; q1_ws[qrow] = s1; }
}

// ---------------------------------------------------------------------------
// Kernel 2: one wave per 16-row strip of one batch. Context is read ONCE:
//   - sp tile (16x64) = (C_rows * w_p) @ Q^T  via V_WMMA_F32_16X16X4_F32
//   - sc = C_row . w_c and ctx1 = C_row . w_in fused into the A-frag stream
//   - per-row masked softmax over j (64 values), U = P @ q1, rowmax
//
// A-frag (16x4 f32): lane l holds row (l&15), K = k0 + 2*(l>>4) + {0,1} -> float2 load
// B-frag (4x16 f32): lane l holds col (l&15), K = k0 + 2*(l>>4) + {0,1} -> float2 load
// C/D (16x16 f32):   lane l, vgpr r -> row r + 8*(l>>4), col (l&15).
// Lanes lm and lm+16 jointly cover all K of row (i_base+lm), so one
// shfl_xor(16) completes the sc/ctx1 dot products; sc for row M is then
// broadcast from lane M into the softmax.
// ---------------------------------------------------------------------------
__global__ void attn_kernel(const float* __restrict__ ctx,
                            const float* __restrict__ qst,
                            const int* __restrict__ mask,
                            const float* __restrict__ att_w,
                            const float* __restrict__ att_b,
                            const float* __restrict__ w_in,
                            const float* __restrict__ sq_ws,
                            const float* __restrict__ q1_ws,
                            float* __restrict__ ctx1_ws,
                            float* __restrict__ U_ws,
                            float* __restrict__ rowmax_ws) {
  const int wid  = (blockIdx.x * blockDim.x + threadIdx.x) >> 5;  // strip id
  const int lane = threadIdx.x & 31;
  const int b      = wid >> 5;          // 32 strips per batch
  const int i_base = (wid & 31) * 16;

  const int half = lane >> 4;   // 0 or 1
  const int lm   = lane & 15;

  const float* wc    = att_w;                                      // w_c
  const float* wp    = att_w + 2 * ND;                             // w_p
  const float* Arow  = ctx + ((size_t)b * LC + i_base + lm) * ND;  // this lane's A row
  const float* Qbase = qst + (size_t)b * LQ * ND;

  v8f acc0 = {}, acc1 = {}, acc2 = {}, acc3 = {};
  float ssc = 0.f;   // partial C_row . w_c   (this lane's half of K)
  float sct = 0.f;   // partial C_row . w_in

  #pragma unroll 4
  for (int k0 = 0; k0 < ND; k0 += 4) {
    const int kk = k0 + 2 * half;
    const float ax = Arow[kk];
    const float ay = Arow[kk + 1];

    ssc += ax * wc[kk]   + ay * wc[kk + 1];
    sct += ax * w_in[kk] + ay * w_in[kk + 1];

    v2f a;
    a.x = ax * wp[kk];
    a.y = ay * wp[kk + 1];

    const float* q0 = Qbase + (size_t)(lm)      * ND + kk;
    const float* q1p = Qbase + (size_t)(16 + lm) * ND + kk;
    const float* q2 = Qbase + (size_t)(32 + lm) * ND + kk;
    const float* q3 = Qbase + (size_t)(48 + lm) * ND + kk;
    v2f b0; b0.x = q0[0];  b0.y = q0[1];
    v2f b1; b1.x = q1p[0]; b1.y = q1p[1];
    v2f b2; b2.x = q2[0];  b2.y = q2[1];
    v2f b3; b3.x = q3[0];  b3.y = q3[1];

    acc0 = __builtin_amdgcn_wmma_f32_16x16x4_f32(false, a, false, b0, (short)0, acc0, false, false);
    acc1 = __builtin_amdgcn_wmma_f32_16x16x4_f32(false, a, false, b1, (short)0, acc1, false, false);
    acc2 = __builtin_amdgcn_wmma_f32_16x16x4_f32(false, a, false, b2, (short)0, acc2, false, false);
    acc3 = __builtin_amdgcn_wmma_f32_16x16x4_f32(false, a, false, b3, (short)0, acc3, false, false);
  }

  // Complete the fused row dot products: lanes lm / lm+16 hold the two
  // K-halves of row (i_base + lm).
  ssc += __shfl_xor(ssc, 16, 32);
  sct += __shfl_xor(sct, 16, 32);
  if (half == 0) ctx1_ws[b * LC + i_base + lm] = sct;   // ctx1, consumed by kernel 3

  // Per-column (j) terms, constant over the 8 rows this lane holds.
  const float bias = att_b[0];
  float sqv[4], q1v[4], mb[4];
  #pragma unroll
  for (int t = 0; t < 4; ++t) {
    const int j = 16 * t + lm;
    sqv[t] = sq_ws[b * LQ + j];
    q1v[t] = q1_ws[b * LQ + j];
    mb[t]  = -NEG_BIG * (1.0f - (float)mask[b * LQ + j]);
  }
  // Per-row sc values: row M = r + 8*half; its dot lives in lane M.
  float sc_r[8];
  #pragma unroll
  for (int r = 0; r < 8; ++r)
    sc_r[r] = __shfl(ssc, r + 8 * half, 32);

  // Row softmax over j: each row lives in one 16-lane half, 4 values per lane.
  #pragma unroll
  for (int r = 0; r < 8; ++r) {
    float v0 = acc0[r] + sc_r[r] + sqv[0] + bias + mb[0];
    float v1 = acc1[r] + sc_r[r] + sqv[1] + bias + mb[1];
    float v2 = acc2[r] + sc_r[r] + sqv[2] + bias + mb[2];
    float v3 = acc3[r] + sc_r[r] + sqv[3] + bias + mb[3];

    float m = fmaxf(fmaxf(v0, v1), fmaxf(v2, v3));
    #pragma unroll
    for (int off = 1; off <= 8; off <<= 1)   // stays within each 16-lane half
      m = fmaxf(m, __shfl_xor(m, off, 32));

    float e0 = __expf(v0 - m), e1 = __expf(v1 - m);
    float e2 = __expf(v2 - m), e3 = __expf(v3 - m);
    float s = e0 + e1 + e2 + e3;
    float u = e0 * q1v[0] + e1 * q1v[1] + e2 * q1v[2] + e3 * q1v[3];
    #pragma unroll
    for (int off = 1; off <= 8; off <<= 1) {
      s += __shfl_xor(s, off, 32);
      u += __shfl_xor(u, off, 32);
    }

    if (lm == 0) {
      const int i = i_base + r + 8 * half;
      U_ws[b * LC + i]      = u / s;
      rowmax_ws[b * LC + i] = m;
    }
  }
}

// ---------------------------------------------------------------------------
// Kernel 3: per-batch softmax over i of rowmax -> scalar H, then fused output
//   G[b,i,:] = [ctx1, U, ctx1*U, U*H]
// One block (256 threads) per batch.
// ---------------------------------------------------------------------------
__global__ void finalize_kernel(const float* __restrict__ ctx1_ws,
                                const float* __restrict__ U_ws,
                                const float* __restrict__ rowmax_ws,
                                float* __restrict__ out) {
  __shared__ float redm[8];
  __shared__ float reds[8];
  __shared__ float redn[8];

  const int b   = blockIdx.x;
  const int tid = threadIdx.x;
  const float* rm = rowmax_ws + b * LC;
  const float* c1 = ctx1_ws + b * LC;

  // global max over 512 row-maxes
  float m = fmaxf(rm[tid], rm[tid + 256]);
  #pragma unroll
  for (int off = 16; off > 0; off >>= 1) m = fmaxf(m, __shfl_xor(m, off, 32));
  if ((tid & 31) == 0) redm[tid >> 5] = m;
  __syncthreads();
  float gmax = redm[0];
  #pragma unroll
  for (int w = 1; w < 8; ++w) gmax = fmaxf(gmax, redm[w]);

  // denom and weighted numerator
  float e0 = __expf(rm[tid] - gmax), e1 = __expf(rm[tid + 256] - gmax);
  float s = e0 + e1;
  float n = e0 * c1[tid] + e1 * c1[tid + 256];
  #pragma unroll
  for (int off = 16; off > 0; off >>= 1) {
    s += __shfl_xor(s, off, 32);
    n += __shfl_xor(n, off, 32);
  }
  if ((tid & 31) == 0) { reds[tid >> 5] = s; redn[tid >> 5] = n; }
  __syncthreads();
  float S = 0.f, N = 0.f;
  #pragma unroll
  for (int w = 0; w < 8; ++w) { S += reds[w]; N += redn[w]; }
  const float H = N / S;

  // fused output, float4 per row
  for (int i = tid; i < LC; i += 256) {
    const float c = c1[i];
    const float u = U_ws[b * LC + i];
    float4 g = make_float4(c, u, c * u, u * H);
    reinterpret_cast<float4*>(out)[b * LC + i] = g;
  }
}

// ---------------------------------------------------------------------------
extern "C" void kernel_launch(void* const* d_in, const int* in_sizes, int n_in,
                              void* d_out, int out_size, void* d_ws, size_t ws_size,
                              hipStream_t stream) {
  const float* ctx   = (const float*)d_in[0];  // (8,512,256)
  const float* qst   = (const float*)d_in[1];  // (8,64,256)
  const int*   mask  = (const int*)d_in[2];    // (8,64)
  const float* att_w = (const float*)d_in[3];  // (768,)
  const float* att_b = (const float*)d_in[4];  // (1,)
  const float* w_in  = (const float*)d_in[5];  // (256,)
  const float* w_mem = (const float*)d_in[6];  // (256,)

  float* ws      = (float*)d_ws;
  float* ctx1_ws = ws;           // 8*512 = 4096
  float* sq_ws   = ws + 4096;    // 8*64 = 512
  float* q1_ws   = ws + 4608;    // 512
  float* U_ws    = ws + 5120;    // 4096
  float* rm_ws   = ws + 9216;    // 4096   (total 13312 floats = 52 KB)

  // Kernel 1: 512 question rows, 4 waves per 128-thread block.
  const int qRows = NB * LQ;                    // 512
  qrow_reduce_kernel<<<(qRows + 3) / 4, 128, 0, stream>>>(qst, att_w, w_mem,
                                                          sq_ws, q1_ws);

  // Kernel 2: 256 strips (8 batches x 32), 4 waves per block -> 64 blocks.
  attn_kernel<<<(NB * LC / 16) / 4, 128, 0, stream>>>(ctx, qst, mask, att_w, att_b,
                                                      w_in, sq_ws, q1_ws,
                                                      ctx1_ws, U_ws, rm_ws);

  // Kernel 3: one block per batch.
  finalize_kernel<<<NB, 256, 0, stream>>>(ctx1_ws, U_ws, rm_ws, (float*)d_out);
}